// MoebiusTransformer_23321672417668
// MI455X (gfx1250) — compile-verified
//
#include <hip/hip_runtime.h>
#include <math.h>

// Möbius transform on 8192 x 512 independent 3-vectors (memory-bound stream).
// Traffic: 144 MiB -> ~6.2 us floor at 23.3 TB/s. WMMA not applicable: the only
// matrix work is a per-vector 3x3 whose determinant is folded to closed form:
//   det(J) = (c1/xn2) * (c2*(x.d)*(d.y) - c1*(x.y))
// via the rank-1 determinant lemma on J = A*B + y (x/xn2)^T (det(A*B)=0 since
// B is a rank-2 projector). Verified on w=0 -> identity map -> det=1.

typedef __attribute__((ext_vector_type(4))) float v4f;

namespace {

constexpr int   kNFeat     = 1536;
constexpr int   kThreads   = 128;       // 4 wave32 waves; 4 vectors (12 floats) per thread
constexpr float kMaxRadius = 0.99f;

// One Möbius 3-vector: writes y0..y2, returns det(J) (analytic, exact value).
__device__ __forceinline__ float moebius_det(float x0, float x1, float x2,
                                             float q0, float q1, float q2,
                                             float& y0, float& y1, float& y2) {
  const float wn2 = q0*q0 + q1*q1 + q2*q2;
  const float wn  = __builtin_amdgcn_sqrtf(wn2);
  const float xn2 = x0*x0 + x1*x1 + x2*x2;
  const float xn  = __builtin_amdgcn_sqrtf(xn2);
  const float r   = xn * (kMaxRadius * __builtin_amdgcn_rcpf(1.0f + wn));

  const float w0 = r*q0, w1 = r*q1, w2 = r*q2;   // rescaled w
  const float wns = r * wn;
  const float num = xn2 - wns*wns;

  const float d0 = x0 - w0, d1 = x1 - w1, d2 = x2 - w2;
  const float dn2     = d0*d0 + d1*d1 + d2*d2;
  const float inv_dn2 = __builtin_amdgcn_rcpf(dn2);
  const float c1      = num * inv_dn2;           // num/dn^2

  y0 = c1*d0 - w0;
  y1 = c1*d1 - w1;
  y2 = c1*d2 - w2;

  const float c2 = 2.0f * c1 * inv_dn2;          // 2*num/dn^4
  const float xy = x0*y0 + x1*y1 + x2*y2;
  const float xd = x0*d0 + x1*d1 + x2*d2;
  const float dy = d0*y0 + d1*y1 + d2*y2;
  const float inv_xn2 = __builtin_amdgcn_rcpf(xn2);

  return (c1 * inv_xn2) * (c2 * xd * dy - c1 * xy);
}

__global__ __launch_bounds__(kThreads)
void moebius_kernel(const float* __restrict__ x,
                    const float* __restrict__ w,
                    float* __restrict__ y,
                    float* __restrict__ logdet) {
  const int    b      = blockIdx.x;   // batch row
  const int    t      = threadIdx.x;  // 0..127
  const size_t rowOff = (size_t)b * kNFeat + (size_t)t * 12;

  const v4f* __restrict__ xq = reinterpret_cast<const v4f*>(x + rowOff);
  const v4f* __restrict__ wq = reinterpret_cast<const v4f*>(w + rowOff);

  // 3 x b128 non-temporal loads per stream (single-use data).
  const v4f a0 = __builtin_nontemporal_load(xq + 0);
  const v4f a1 = __builtin_nontemporal_load(xq + 1);
  const v4f a2 = __builtin_nontemporal_load(xq + 2);
  const v4f b0 = __builtin_nontemporal_load(wq + 0);
  const v4f b1 = __builtin_nontemporal_load(wq + 1);
  const v4f b2 = __builtin_nontemporal_load(wq + 2);

  // 4 vectors per thread, components consumed straight from load registers.
  float y00, y01, y02, y10, y11, y12, y20, y21, y22, y30, y31, y32;
  float prod;
  prod  = moebius_det(a0.x, a0.y, a0.z,  b0.x, b0.y, b0.z,  y00, y01, y02);
  prod *= moebius_det(a0.w, a1.x, a1.y,  b0.w, b1.x, b1.y,  y10, y11, y12);
  prod *= moebius_det(a1.z, a1.w, a2.x,  b1.z, b1.w, b2.x,  y20, y21, y22);
  prod *= moebius_det(a2.y, a2.z, a2.w,  b2.y, b2.z, b2.w,  y30, y31, y32);

  // sum(log|det_i|) == log|prod det_i|: one v_log_f32 per thread instead of 4.
  float sum = __logf(fabsf(prod));

  v4f* __restrict__ yq = reinterpret_cast<v4f*>(y + rowOff);
  const v4f o0 = {y00, y01, y02, y10};
  const v4f o1 = {y11, y12, y20, y21};
  const v4f o2 = {y22, y30, y31, y32};
  __builtin_nontemporal_store(o0, yq + 0);
  __builtin_nontemporal_store(o1, yq + 1);
  __builtin_nontemporal_store(o2, yq + 2);

  // Deterministic wave32 shuffle-tree reduction, then LDS combine of 4 waves.
#pragma unroll
  for (int off = 16; off > 0; off >>= 1) {
    sum += __shfl_down(sum, off, 32);
  }
  __shared__ float red[kThreads / 32];
  if ((t & 31) == 0) red[t >> 5] = sum;
  __syncthreads();
  if (t == 0) {
    float s = 0.0f;
#pragma unroll
    for (int i = 0; i < kThreads / 32; ++i) s += red[i];
    logdet[b] = s;
  }
}

} // namespace

extern "C" void kernel_launch(void* const* d_in, const int* in_sizes, int n_in,
                              void* d_out, int out_size, void* d_ws, size_t ws_size,
                              hipStream_t stream) {
  (void)n_in; (void)out_size; (void)d_ws; (void)ws_size;
  const float* x = (const float*)d_in[0];          // (B, 1536)
  const float* w = (const float*)d_in[1];          // (B, 1, 1536) — same flat layout
  const int batch = in_sizes[0] / kNFeat;          // 8192

  float* y      = (float*)d_out;                   // (B, 1536)
  float* logdet = (float*)d_out + (size_t)batch * kNFeat; // (B,)

  moebius_kernel<<<dim3(batch), dim3(kThreads), 0, stream>>>(x, w, y, logdet);
}